// SSDCriterion_33956011442370
// MI455X (gfx1250) — compile-verified
//
#include <hip/hip_runtime.h>
#include <stdint.h>

#define TPB        256
#define NBLK       2048
#define CH         81
#define TILE       64                      // anchors per tile
#define TILE_F     (TILE * CH)             // 5184 floats
#define TILE_V4    (TILE_F / 4)            // 1296 b128 transfers
#define TILE_BYTES (TILE_F * 4)            // 20736 B per buffer

__device__ __forceinline__ float smooth_l1(float d) {
    float ad = fabsf(d);
    return ad < 1.0f ? 0.5f * d * d : ad - 0.5f;
}

__global__ __launch_bounds__(TPB) void ssd_partials_kernel(
    const float* __restrict__ locP, const float* __restrict__ locT,
    const float* __restrict__ clsP, const int* __restrict__ tgt,
    float* __restrict__ partials, int nAnchors, int nTiles, int nVec4Total)
{
    __shared__ float sh[2 * TILE_F];       // 41472 B double buffer
    __shared__ float red[3][TPB / 32];

    const int tid = threadIdx.x;
    const int g   = tid >> 2;              // anchor group within tile (0..63)
    const int r   = tid & 3;               // lane role within group

    // Low 32 bits of a generic pointer to LDS == LDS byte offset.
    const uint32_t shbase = (uint32_t)(uint64_t)(uintptr_t)(&sh[0]);
    const uint64_t sbase  = (uint64_t)(uintptr_t)clsP;

    float locAcc = 0.0f, ceAcc = 0.0f;
    int   posAcc = 0;

    // Issue async DMA of one tile's class logits into LDS buffer `buf`.
    auto issue = [&](int buf, int tile) {
        const int kbase = tile * TILE_V4;
        const int kmax  = min(TILE_V4, nVec4Total - kbase);
        const uint32_t ldsbuf = shbase + (uint32_t)buf * TILE_BYTES;
        for (int k = tid; k < kmax; k += TPB) {
            uint32_t lds_addr = ldsbuf + (uint32_t)k * 16u;
            uint32_t goff     = (uint32_t)((kbase + k) * 16);   // < 255 MB, fits i32
            asm volatile("global_load_async_to_lds_b128 %0, %1, %2"
                         :: "v"(lds_addr), "v"(goff), "s"(sbase) : "memory");
        }
    };

    int t   = (int)blockIdx.x;
    int buf = 0;
    if (t < nTiles) issue(0, t);

    for (; t < nTiles; t += (int)gridDim.x) {
        // Wait for this wave's DMA, then make all waves' LDS writes visible.
        asm volatile("s_wait_asynccnt 0x0" ::: "memory");
        __syncthreads();

        int tn = t + (int)gridDim.x;
        if (tn < nTiles) issue(buf ^ 1, tn);   // overlap next tile's DMA

        const float* shb = sh + buf * TILE_F;
        const int a = t * TILE + g;
        if (a < nAnchors) {
            // 4-lane cooperative log-sum-exp over 81 channels
            float m = -3.402823466e38f;
            for (int c = r; c < CH; c += 4) m = fmaxf(m, shb[g * CH + c]);
            m = fmaxf(m, __shfl_xor(m, 1, 32));
            m = fmaxf(m, __shfl_xor(m, 2, 32));
            float s = 0.0f;
            for (int c = r; c < CH; c += 4) s += __expf(shb[g * CH + c] - m);
            s += __shfl_xor(s, 1, 32);
            s += __shfl_xor(s, 2, 32);

            if (r == 0) {
                const int tg = tgt[a];
                if (tg != 0) {                 // positive anchor
                    posAcc += 1;
                    const float lse = m + __logf(s);
                    ceAcc += lse - shb[g * CH + tg];

                    const float4 lp = reinterpret_cast<const float4*>(locP)[a];
                    const float4 lt = reinterpret_cast<const float4*>(locT)[a];
                    locAcc += smooth_l1(lp.x - lt.x) + smooth_l1(lp.y - lt.y)
                            + smooth_l1(lp.z - lt.z) + smooth_l1(lp.w - lt.w);
                }
            }
        }
        __syncthreads();                       // all reads of `buf` done
        buf ^= 1;
    }

    // Deterministic block reduction (wave32 shuffles + LDS)
    float pAcc = (float)posAcc;
    for (int off = 16; off > 0; off >>= 1) {
        locAcc += __shfl_down(locAcc, off, 32);
        ceAcc  += __shfl_down(ceAcc,  off, 32);
        pAcc   += __shfl_down(pAcc,   off, 32);
    }
    const int wave = tid >> 5, lane = tid & 31;
    __syncthreads();
    if (lane == 0) { red[0][wave] = locAcc; red[1][wave] = ceAcc; red[2][wave] = pAcc; }
    __syncthreads();
    if (tid == 0) {
        float L = 0.0f, C = 0.0f, P = 0.0f;
        for (int w = 0; w < TPB / 32; ++w) { L += red[0][w]; C += red[1][w]; P += red[2][w]; }
        partials[blockIdx.x]            = L;
        partials[NBLK + blockIdx.x]     = C;
        partials[2 * NBLK + blockIdx.x] = P;
    }
}

__global__ __launch_bounds__(TPB) void ssd_finalize_kernel(
    const float* __restrict__ partials, float* __restrict__ out)
{
    __shared__ float red[3][TPB];
    const int tid = threadIdx.x;
    float L = 0.0f, C = 0.0f, P = 0.0f;
    for (int i = tid; i < NBLK; i += TPB) {
        L += partials[i];
        C += partials[NBLK + i];
        P += partials[2 * NBLK + i];
    }
    red[0][tid] = L; red[1][tid] = C; red[2][tid] = P;
    __syncthreads();
    for (int s = TPB / 2; s > 0; s >>= 1) {
        if (tid < s) {
            red[0][tid] += red[0][tid + s];
            red[1][tid] += red[1][tid + s];
            red[2][tid] += red[2][tid + s];
        }
        __syncthreads();
    }
    if (tid == 0) {
        const float np = red[2][0];
        out[0] = red[0][0] / np;   // loc_loss / num_pos
        out[1] = red[1][0] / np;   // cls_loss / num_pos
    }
}

extern "C" void kernel_launch(void* const* d_in, const int* in_sizes, int n_in,
                              void* d_out, int out_size, void* d_ws, size_t ws_size,
                              hipStream_t stream) {
    (void)n_in; (void)out_size; (void)ws_size;
    const float* locP = (const float*)d_in[0];   // (B, M, 4) f32
    const float* locT = (const float*)d_in[1];   // (B, M, 4) f32
    const float* clsP = (const float*)d_in[2];   // (B, M, C) f32
    const int*   tgt  = (const int*)d_in[3];     // (B, M) int

    const int nAnchors   = in_sizes[3];                    // B*M
    const int nTiles     = (nAnchors + TILE - 1) / TILE;
    const int nVec4Total = (nAnchors * CH) / 4;            // 81*B*M divisible by 4

    float* partials = (float*)d_ws;                        // 3*NBLK floats (24 KB)
    float* out      = (float*)d_out;

    ssd_partials_kernel<<<dim3(NBLK), dim3(TPB), 0, stream>>>(
        locP, locT, clsP, tgt, partials, nAnchors, nTiles, nVec4Total);
    ssd_finalize_kernel<<<dim3(1), dim3(TPB), 0, stream>>>(partials, out);
}